// DMLoss_2705829396669
// MI455X (gfx1250) — compile-verified
//
#include <hip/hip_runtime.h>
#include <hip/hip_bf16.h>

// DMLoss for MI455X (gfx1250, wave32).
// Item-1 pairwise distances via V_WMMA_F32_16X16X4_F32 with the duplicated-K trick:
//   A (16x4): cols K0,K1 = (px,py), cols K2,K3 = (px,py) again (lanes 16-31)
//   B (4x16): rows K0,K1 = (tx,ty), rows K2,K3 = (tx,ty) again (lanes 16-31)
//   => D = A*B = 2*(p . t),  so  ||p-t||^2 = |p|^2 + |t|^2 - D.  No zero-padding,
//   no predicated loads, EXEC all-ones by construction.

#define NP        128          // points per contour (P)
#define TSTEPS    10           // TIME
#define NT        (NP*TSTEPS)  // 1280 interpolated targets
#define BETA_C    0.25f        // 1/STRIDE
#define BOUND2_C  1000000.0f
#define BIGF      3.4e38f

typedef __attribute__((ext_vector_type(2))) float v2f;
typedef __attribute__((ext_vector_type(4))) float v4f;
typedef __attribute__((ext_vector_type(8))) float v8f;

__device__ __forceinline__ float smooth_l1(float pred, float target) {
    float d = fabsf(pred - target);
    // BETA = 0.25: 0.5*d*d/0.25 = 2*d*d (exact, /0.25 is a power-of-two scale)
    return (d < BETA_C) ? (0.5f * d * d / BETA_C) : (d - 0.5f * BETA_C);
}

__global__ __launch_bounds__(256)
void dmloss_main_kernel(const float* __restrict__ predC,
                        const float* __restrict__ predO,
                        const float* __restrict__ gtC,
                        const float* __restrict__ gtK,
                        const int*   __restrict__ kmask,
                        float*       __restrict__ ws) {
    __shared__ v4f   sT4[NT];        // {tx, ty, |t|^2, 0}  -> one ds_load_b128 per tile
    __shared__ v4f   sP4[NP];        // {px, py, |p|^2, 0}
    __shared__ float sOx[NP], sOy[NP];
    __shared__ float sBD[NP];        // best dist per pred row
    __shared__ int   sBJ[NP];        // best target idx per pred row
    __shared__ float sW[8 * 4];      // per-wave partials

    const int n   = blockIdx.x;
    const int tid = threadIdx.x;

    const float* gc = gtC   + (size_t)n * NP * 2;
    const float* pc = predC + (size_t)n * NP * 2;
    const float* po = predO + (size_t)n * NP * 2;
    const float* gk = gtK   + (size_t)n * NP * 2;

    if (tid == 0) {
        __builtin_prefetch(gc, 0, 3);   // global_prefetch_b8
        __builtin_prefetch(pc, 0, 3);
        __builtin_prefetch(po, 0, 3);
        __builtin_prefetch(gk, 0, 3);
    }

    // ---- stage preds into LDS ----
    if (tid < NP) {
        float x = pc[2 * tid], y = pc[2 * tid + 1];
        v4f pv; pv.x = x; pv.y = y; pv.z = x * x + y * y; pv.w = 0.0f;
        sP4[tid] = pv;
        sOx[tid] = po[2 * tid]; sOy[tid] = po[2 * tid + 1];
    }
    // ---- build interpolated targets: tgt[p*10+t] = poly[p]*t/10 + poly[p-1]*(1-t/10) ----
    for (int j = tid; j < NT; j += 256) {
        int p = j / TSTEPS, t = j - p * TSTEPS;
        float step = (float)t / (float)TSTEPS;
        int pm = (p == 0) ? (NP - 1) : (p - 1);
        float gx = gc[2 * p],  gy = gc[2 * p + 1];
        float hx = gc[2 * pm], hy = gc[2 * pm + 1];
        float txv = gx * step + hx * (1.0f - step);
        float tyv = gy * step + hy * (1.0f - step);
        v4f tv; tv.x = txv; tv.y = tyv; tv.z = txv * txv + tyv * tyv; tv.w = 0.0f;
        sT4[j] = tv;
    }
    __syncthreads();

    // ---- item 1: WMMA nearest-target search, one wave per 16 pred rows ----
    const int wave    = tid >> 5;
    const int lane    = tid & 31;
    const int half    = lane >> 4;
    const int lm      = lane & 15;
    const int rowBase = wave * 16;

    // A fragment: every lane carries (x,y) of its row; lanes 16-31 duplicate as K2,K3
    v4f pself = sP4[rowBase + lm];
    v2f afrag; afrag.x = pself.x; afrag.y = pself.y;

    float ppg[8];
    #pragma unroll
    for (int g = 0; g < 8; ++g) ppg[g] = sP4[rowBase + g + 8 * half].z;

    float best[8]; int bidx[8];
    #pragma unroll
    for (int g = 0; g < 8; ++g) { best[g] = BIGF; bidx[g] = 0; }

    #pragma unroll 2
    for (int tile = 0; tile < NT / 16; ++tile) {
        const int j0 = tile * 16 + lm;
        v4f tv = sT4[j0];                       // single ds_load_b128
        v2f bfrag; bfrag.x = tv.x; bfrag.y = tv.y;

        v8f c = {};
        c = __builtin_amdgcn_wmma_f32_16x16x4_f32(
                /*neg_a=*/false, afrag, /*neg_b=*/false, bfrag,
                /*c_mod=*/(short)0, c, /*reuse_a=*/false, /*reuse_b=*/false);

        // c[g] = 2 * dot(pred row M=g+8*half, target col N=lm)
        #pragma unroll
        for (int g = 0; g < 8; ++g) {
            float d = (ppg[g] + tv.z) - c[g];
            if (d < best[g]) { best[g] = d; bidx[g] = j0; }
        }
    }

    // min/argmin across the 16 lanes of each half (xor masks < 16 stay within a half)
    #pragma unroll
    for (int g = 0; g < 8; ++g) {
        float bd = best[g]; int bj = bidx[g];
        #pragma unroll
        for (int m = 8; m >= 1; m >>= 1) {
            float od = __shfl_xor(bd, m, 32);
            int   oj = __shfl_xor(bj, m, 32);
            if (od < bd) { bd = od; bj = oj; }
        }
        if (lm == 0) {
            sBD[rowBase + g + 8 * half] = bd;
            sBJ[rowBase + g + 8 * half] = bj;
        }
    }
    __syncthreads();

    // ---- per-point loss terms ----
    float s1 = 0.0f, c1 = 0.0f, s2 = 0.0f, c2 = 0.0f;
    if (tid < NP) {
        // item 1 term for pred point tid
        float bd = sBD[tid];
        if (bd <= BOUND2_C) {
            int j = sBJ[tid];
            v4f tj = sT4[j];
            v4f pi = sP4[tid];
            float otx = (tj.x - pi.x) * 0.25f;   // /STRIDE
            float oty = (tj.y - pi.y) * 0.25f;
            s1 = smooth_l1(sOx[tid], otx) + smooth_l1(sOy[tid], oty);
            c1 = 1.0f;
        }
        // item 2: key point tid vs 128 preds (tiny, scalar scan; first-min == argmin tie rule)
        float kx = gk[2 * tid], ky = gk[2 * tid + 1];
        int   mk = kmask[(size_t)n * NP + tid];
        float bd2 = BIGF; int bj2 = 0;
        #pragma unroll 4
        for (int j = 0; j < NP; ++j) {
            float dx = kx - sP4[j].x, dy = ky - sP4[j].y;
            float d = dx * dx + dy * dy;
            if (d < bd2) { bd2 = d; bj2 = j; }
        }
        if ((bd2 <= BOUND2_C) && mk) {
            float otx = (kx - sP4[bj2].x) * 0.25f;
            float oty = (ky - sP4[bj2].y) * 0.25f;
            s2 = smooth_l1(sOx[bj2], otx) + smooth_l1(sOy[bj2], oty);
            c2 = 1.0f;
        }
    }

    // ---- deterministic block reduction: wave shfl-sum -> LDS -> thread 0 ----
    #pragma unroll
    for (int m = 16; m >= 1; m >>= 1) {
        s1 += __shfl_xor(s1, m, 32);
        c1 += __shfl_xor(c1, m, 32);
        s2 += __shfl_xor(s2, m, 32);
        c2 += __shfl_xor(c2, m, 32);
    }
    if (lane == 0) {
        sW[wave * 4 + 0] = s1; sW[wave * 4 + 1] = c1;
        sW[wave * 4 + 2] = s2; sW[wave * 4 + 3] = c2;
    }
    __syncthreads();
    if (tid == 0) {
        float a0 = 0.f, a1 = 0.f, a2 = 0.f, a3 = 0.f;
        #pragma unroll
        for (int w = 0; w < 8; ++w) {
            a0 += sW[4 * w + 0]; a1 += sW[4 * w + 1];
            a2 += sW[4 * w + 2]; a3 += sW[4 * w + 3];
        }
        ws[4 * n + 0] = a0; ws[4 * n + 1] = a1;
        ws[4 * n + 2] = a2; ws[4 * n + 3] = a3;
    }
}

__global__ void dmloss_final_kernel(const float* __restrict__ ws, float* __restrict__ out, int nblk) {
    if (threadIdx.x == 0 && blockIdx.x == 0) {
        float s1 = 0.f, c1 = 0.f, s2 = 0.f, c2 = 0.f;
        for (int n = 0; n < nblk; ++n) {          // fixed order -> deterministic
            s1 += ws[4 * n + 0]; c1 += ws[4 * n + 1];
            s2 += ws[4 * n + 2]; c2 += ws[4 * n + 3];
        }
        float l1 = (s1 / fmaxf(2.0f * c1, 1.0f)) * 0.5f;  // * (1 - KEY_W)
        float l2 = (s2 / fmaxf(2.0f * c2, 1.0f)) * 0.5f;  // * KEY_W
        out[0] = l1 + l2;                                  // * LOSS_WEIGHT (1.0)
    }
}

extern "C" void kernel_launch(void* const* d_in, const int* in_sizes, int n_in,
                              void* d_out, int out_size, void* d_ws, size_t ws_size,
                              hipStream_t stream) {
    const float* predC = (const float*)d_in[0];  // pred_contours (N,P,2)
    const float* predO = (const float*)d_in[1];  // pred_offsets  (N,P,2)
    const float* gtC   = (const float*)d_in[2];  // gt_contours   (N,P,2)
    const float* gtK   = (const float*)d_in[3];  // gt_key_points (N,P,2)
    const int*   kmask = (const int*)d_in[4];    // gt_key_points_mask (N,P)
    float*       out   = (float*)d_out;
    float*       ws    = (float*)d_ws;           // 4 floats per contour

    const int N = in_sizes[0] / (NP * 2);        // 256 for the reference shapes

    dmloss_main_kernel<<<N, 256, 0, stream>>>(predC, predO, gtC, gtK, kmask, ws);
    dmloss_final_kernel<<<1, 32, 0, stream>>>(ws, out, N);
}